// AVWGCN_74354473828677
// MI455X (gfx1250) — compile-verified
//
#include <hip/hip_runtime.h>
#include <math.h>

typedef __attribute__((ext_vector_type(2))) float v2f;
typedef __attribute__((ext_vector_type(8))) float v8f;

#define Nn 4096
#define Dd 16
#define Bb 64
#define Cc 32
#define Oo 32

// ---------------------------------------------------------------------------
// Kernel 1: A = row_softmax(relu(E E^T)), one block per row.
// ---------------------------------------------------------------------------
__global__ __launch_bounds__(256) void build_adj(const float* __restrict__ E,
                                                 float* __restrict__ A) {
    __shared__ float vals[Nn];
    __shared__ float red[256];
    __shared__ float en[Dd];
    const int n = blockIdx.x;
    const int t = threadIdx.x;
    if (t < Dd) en[t] = E[n * Dd + t];
    __syncthreads();

    float lmax = 0.0f;
    for (int m = t; m < Nn; m += 256) {
        const float4* ep = (const float4*)(E + (size_t)m * Dd);
        float4 e0 = ep[0], e1 = ep[1], e2 = ep[2], e3 = ep[3];
        float dot = e0.x*en[0]  + e0.y*en[1]  + e0.z*en[2]  + e0.w*en[3]
                  + e1.x*en[4]  + e1.y*en[5]  + e1.z*en[6]  + e1.w*en[7]
                  + e2.x*en[8]  + e2.y*en[9]  + e2.z*en[10] + e2.w*en[11]
                  + e3.x*en[12] + e3.y*en[13] + e3.z*en[14] + e3.w*en[15];
        float v = dot > 0.0f ? dot : 0.0f;   // relu
        vals[m] = v;
        lmax = fmaxf(lmax, v);
    }
    red[t] = lmax;
    __syncthreads();
    for (int s = 128; s > 0; s >>= 1) {
        if (t < s) red[t] = fmaxf(red[t], red[t + s]);
        __syncthreads();
    }
    const float rmax = red[0];
    __syncthreads();

    float lsum = 0.0f;
    for (int m = t; m < Nn; m += 256) {
        float ex = __expf(vals[m] - rmax);
        vals[m] = ex;
        lsum += ex;
    }
    red[t] = lsum;
    __syncthreads();
    for (int s = 128; s > 0; s >>= 1) {
        if (t < s) red[t] += red[t + s];
        __syncthreads();
    }
    const float inv = 1.0f / red[0];
    for (int m = t; m < Nn; m += 256) A[(size_t)n * Nn + m] = vals[m] * inv;
}

// ---------------------------------------------------------------------------
// Kernel 2/3: Z[b] = alpha * A @ S[b] + beta * X0[b]
// A: [N,N] f32, S/X0/Z: [B,N,C] f32.  256 threads = 8 waves.
// Block tile: 128 rows x 32 cols; wave tile: 16 x 32 (two v8f accumulators).
// K staged through LDS in chunks of 32; X chunk is staged TRANSPOSED so both
// A and B fragments are contiguous 64-bit LDS loads feeding
// v_wmma_f32_16x16x4_f32.  Next A chunk is prefetched (global_prefetch_b8).
// ---------------------------------------------------------------------------
__global__ __launch_bounds__(256) void agg_gemm(const float* __restrict__ A,
                                                const float* __restrict__ S,
                                                const float* __restrict__ X0,
                                                float* __restrict__ Z,
                                                float alpha, float beta) {
    __shared__ float As [128][36];  // [row][k]  (+4 pad: alignment + bank spread)
    __shared__ float XsT[32][36];   // [col][k]  transposed X chunk

    const int t       = threadIdx.x;
    const int rowBase = blockIdx.x * 128;
    const int b       = blockIdx.y;
    const float* Sb   = S  + (size_t)b * Nn * Cc;
    const float* Xb   = X0 + (size_t)b * Nn * Cc;
    float*       Zb   = Z  + (size_t)b * Nn * Cc;

    const int wave  = t >> 5;        // 0..7
    const int lane  = t & 31;
    const int half  = lane >> 4;     // 0: K pair {0,1}, 1: K pair {2,3}
    const int mr    = lane & 15;
    const int waveM = wave * 16;

    v8f acc0 = {};
    v8f acc1 = {};

    for (int k0 = 0; k0 < Nn; k0 += 32) {
        // Stage A chunk: 128 x 32 floats (1024 float4, 4 per thread),
        // and prefetch the next K-chunk of A into cache.
        #pragma unroll
        for (int i = 0; i < 4; ++i) {
            int q  = t + i * 256;          // 0..1023
            int r  = q >> 3;               // row 0..127
            int c4 = q & 7;                // float4 column 0..7
            const float* gp = A + (size_t)(rowBase + r) * Nn + k0 + c4 * 4;
            float4 v = *(const float4*)gp;
            *(float4*)(&As[r][c4 * 4]) = v;
            if (k0 + 32 < Nn) __builtin_prefetch(gp + 32, 0, 3);
        }
        // Stage X chunk transposed: 32 x 32 floats -> XsT[col][k]
        {
            int r  = t >> 3;               // k row 0..31
            int c4 = t & 7;
            float4 v = *(const float4*)(Sb + (size_t)(k0 + r) * Cc + c4 * 4);
            XsT[c4 * 4 + 0][r] = v.x;
            XsT[c4 * 4 + 1][r] = v.y;
            XsT[c4 * 4 + 2][r] = v.z;
            XsT[c4 * 4 + 3][r] = v.w;
        }
        __syncthreads();

        #pragma unroll
        for (int s = 0; s < 8; ++s) {
            const int ka = s * 4 + half * 2;
            // A fragment (16x4): lanes 0-15 rows M, K={0,1}; lanes 16-31 K={2,3}
            v2f af = *(const v2f*)(&As[waveM + mr][ka]);
            // B fragments (4x16): lanes 0-15 cols N, K={0,1}; lanes 16-31 K={2,3}
            v2f b0 = *(const v2f*)(&XsT[mr][ka]);
            v2f b1 = *(const v2f*)(&XsT[16 + mr][ka]);
            acc0 = __builtin_amdgcn_wmma_f32_16x16x4_f32(false, af, false, b0,
                                                         (short)0, acc0, false, false);
            acc1 = __builtin_amdgcn_wmma_f32_16x16x4_f32(false, af, false, b1,
                                                         (short)0, acc1, false, false);
        }
        __syncthreads();
    }

    // D layout (16x16 f32): VGPR i -> row (i + half*8), col mr (+16 for tile 1)
    #pragma unroll
    for (int i = 0; i < 8; ++i) {
        const int row = rowBase + waveM + half * 8 + i;
        {
            const int col = mr;
            Zb[(size_t)row * Cc + col] =
                alpha * acc0[i] + beta * Xb[(size_t)row * Cc + col];
        }
        {
            const int col = 16 + mr;
            Zb[(size_t)row * Cc + col] =
                alpha * acc1[i] + beta * Xb[(size_t)row * Cc + col];
        }
    }
}

// ---------------------------------------------------------------------------
// Kernel 4: per-node adaptive weights + contraction + bias.
// One block per node n: W[n] = sum_d E[n,d]*Wp[d] built in LDS (12 KB),
// then out[b,n,o] = bias[n,o] + sum_c x*W0 + y1*W1 + y2*W2 over all 64 batches.
// ---------------------------------------------------------------------------
__global__ __launch_bounds__(256) void out_kernel(const float* __restrict__ x,
                                                  const float* __restrict__ y1,
                                                  const float* __restrict__ y2,
                                                  const float* __restrict__ E,
                                                  const float* __restrict__ Wp,
                                                  const float* __restrict__ bp,
                                                  float* __restrict__ out) {
    __shared__ float Wsh[3 * Cc * Oo];   // [k][c][o] = 3072 floats
    __shared__ float bsh[Oo];
    __shared__ float en[Dd];
    const int n = blockIdx.x;
    const int t = threadIdx.x;
    if (t < Dd) en[t] = E[n * Dd + t];
    __syncthreads();

    for (int j = t; j < 3 * Cc * Oo; j += 256) {
        float acc = 0.0f;
        #pragma unroll
        for (int d = 0; d < Dd; ++d) acc += en[d] * Wp[d * (3 * Cc * Oo) + j];
        Wsh[j] = acc;
    }
    if (t < Oo) {
        float acc = 0.0f;
        #pragma unroll
        for (int d = 0; d < Dd; ++d) acc += en[d] * bp[d * Oo + t];
        bsh[t] = acc;
    }
    __syncthreads();

    const int o   = t & 31;
    const int blo = t >> 5;              // 0..7
    for (int bb = 0; bb < 8; ++bb) {
        const int b = bb * 8 + blo;
        const size_t base = ((size_t)b * Nn + n) * Cc;
        float acc = bsh[o];
        #pragma unroll
        for (int c = 0; c < Cc; ++c) {
            acc += x [base + c] * Wsh[              c * Oo + o];
            acc += y1[base + c] * Wsh[Cc * Oo     + c * Oo + o];
            acc += y2[base + c] * Wsh[2 * Cc * Oo + c * Oo + o];
        }
        out[base + o] = acc;
    }
}

// ---------------------------------------------------------------------------
extern "C" void kernel_launch(void* const* d_in, const int* in_sizes, int n_in,
                              void* d_out, int out_size, void* d_ws, size_t ws_size,
                              hipStream_t stream) {
    const float* x  = (const float*)d_in[0];   // [B,N,C]
    const float* E  = (const float*)d_in[1];   // [N,D]
    const float* Wp = (const float*)d_in[2];   // [D,K,C,O]
    const float* bp = (const float*)d_in[3];   // [D,O]
    float* out = (float*)d_out;                // [B,N,O]

    float* A  = (float*)d_ws;                  // [N,N]      64 MB
    float* y1 = A  + (size_t)Nn * Nn;          // [B,N,C]    32 MB
    float* y2 = y1 + (size_t)Bb * Nn * Cc;     // [B,N,C]    32 MB

    build_adj<<<Nn, 256, 0, stream>>>(E, A);
    // y1 = A @ x        (Chebyshev T1 term)
    agg_gemm<<<dim3(Nn / 128, Bb), 256, 0, stream>>>(A, x,  x, y1, 1.0f,  0.0f);
    // y2 = 2*A @ y1 - x (Chebyshev T2 term, avoids materializing A@A)
    agg_gemm<<<dim3(Nn / 128, Bb), 256, 0, stream>>>(A, y1, x, y2, 2.0f, -1.0f);
    out_kernel<<<Nn, 256, 0, stream>>>(x, y1, y2, E, Wp, bp, out);
}